// MinimumSpanningMT3DTree_46978352283940
// MI455X (gfx1250) — compile-verified
//
#include <hip/hip_runtime.h>
#include <hip/hip_bf16.h>

// ---------------------------------------------------------------------------
// MinimumSpanningMT3DTree for MI455X (gfx1250, wave32)
// Phases: build edge index -> node norms -> edge cosine weights (f32 WMMA)
//         -> 17 Boruvka rounds (compress / scatter-min / select / relink)
//         -> ordered stream compaction of selected edges.
// ---------------------------------------------------------------------------

namespace {

constexpr int   Bn        = 4;
constexpr int   Cn        = 64;
constexpr int   Hn        = 192;
constexpr int   Wn        = 576;
constexpr int   MIDn      = 192;                 // Wn / 3
constexpr int   Vn        = Hn * Wn;             // 110592
constexpr int   NRt       = (Hn - 1) * MIDn;     // 36672 row edges per tile
constexpr int   NCt       = Hn * (MIDn - 1);     // 36672 col edges per tile
constexpr int   TBLK      = NRt + NCt;           // 73344 edges per tile
constexpr int   CROSSBASE = 3 * TBLK;            // 220032
constexpr int   NCROSS    = Hn * MIDn;           // 36864
constexpr int   NEDGE     = CROSSBASE + NCROSS;  // 256896 actual edges
constexpr int   BIGE      = 6 * TBLK + NCROSS;   // 476928 = reference sentinel E
constexpr int   NOUT      = Vn - 2;              // 110590 outputs per batch
constexpr int   ITERS     = 17;                  // ceil(log2(V))
constexpr float EPSV      = 1e-8f;
constexpr unsigned ENC_INF = 0xFF800000u;        // fkey(+inf)
constexpr int   NBC       = (NEDGE + 1023) / 1024; // 251 compaction blocks

typedef float v2f __attribute__((ext_vector_type(2)));
typedef float v8f __attribute__((ext_vector_type(8)));

// Order-preserving float -> u32 key (for atomicMin), and inverse.
__device__ __forceinline__ unsigned fkey(float f) {
  unsigned u = __float_as_uint(f);
  return (u & 0x80000000u) ? ~u : (u | 0x80000000u);
}
__device__ __forceinline__ float fdecode(unsigned k) {
  unsigned u = (k & 0x80000000u) ? (k ^ 0x80000000u) : ~k;
  return __uint_as_float(u);
}

// --------------------------- graph construction ----------------------------

__global__ void k_build_index(int* __restrict__ eu, int* __restrict__ ev) {
  int e = blockIdx.x * 256 + threadIdx.x;
  if (e >= NEDGE) return;
  int u, v;
  if (e < CROSSBASE) {
    int t = e / TBLK, r = e % TBLK, cb = t * MIDn;
    if (r < NRt) {                       // row edge: (h,w)-(h+1,w)
      int h = r / MIDn, w = r % MIDn;
      u = h * Wn + cb + w;
      v = (h + 1) * Wn + cb + w;
    } else {                             // col edge: (h,w)-(h,w+1)
      int i = r - NRt;
      int h = i / (MIDn - 1), w = i % (MIDn - 1);
      u = h * Wn + cb + w;
      v = u + 1;
    }
  } else {                               // cross: T1(h,w)-T2(h,w)
    int i = e - CROSSBASE;
    int h = i / MIDn, w = i % MIDn;
    u = h * Wn + w;
    v = h * Wn + MIDn + w;
  }
  eu[e] = u;
  ev[e] = v;
}

__global__ void k_node_norm(const float* __restrict__ feat, float* __restrict__ nrm) {
  int n = blockIdx.x * 256 + threadIdx.x;
  int b = blockIdx.y;
  if (n >= Vn) return;
  const float* fb = feat + (size_t)b * Cn * Vn + n;
  float s = 0.f;
#pragma unroll 8
  for (int c = 0; c < Cn; ++c) {
    float x = fb[(size_t)c * Vn];
    s = fmaf(x, x, s);
  }
  nrm[(size_t)b * Vn + n] = fmaxf(sqrtf(s), EPSV);
}

// Edge dot products via f32 WMMA: one wave handles 16 edges; 16x16x4 WMMA
// accumulated over K=64 channels; the 16x16 result's DIAGONAL holds the
// per-edge dots (full f32 precision, same as the reference).
__global__ void k_edge_weight_wmma(const float* __restrict__ feat,
                                   const int* __restrict__ eu,
                                   const int* __restrict__ ev,
                                   const float* __restrict__ nrm,
                                   float* __restrict__ weight) {
  const int b    = blockIdx.y;
  const int lane = threadIdx.x & 31;
  const int wv   = threadIdx.x >> 5;
  const int e0   = (blockIdx.x * 8 + wv) * 16;   // grid sized exactly

  const int row  = (lane < 16) ? lane : (lane - 16);
  const int e    = e0 + row;
  const int un   = eu[e];
  const int vn   = ev[e];
  const int cofs = (lane < 16) ? 0 : 2;          // lanes 16-31 hold K=2,3

  const float* fb = feat + (size_t)b * Cn * Vn;
  v8f acc = {};
#pragma unroll 4
  for (int c0 = 0; c0 < Cn; c0 += 4) {
    const float* fu = fb + (size_t)(c0 + cofs) * Vn;
    const float* fv = fb + (size_t)(c0 + cofs) * Vn;
    v2f a, bm;
    a.x  = fu[un];
    a.y  = fu[(size_t)Vn + un];
    bm.x = fv[vn];
    bm.y = fv[(size_t)Vn + vn];
    // v_wmma_f32_16x16x4_f32: D = A(16x4) x B(4x16) + C  (all f32)
    acc = __builtin_amdgcn_wmma_f32_16x16x4_f32(
        /*neg_a=*/false, a, /*neg_b=*/false, bm,
        /*c_mod=*/(short)0, acc, /*reuse_a=*/false, /*reuse_b=*/false);
  }

  // Diagonal (i,i): lanes 0-7 hold i=0..7 in vgpr=lane; lanes 24-31 hold
  // i=8..15 in vgpr=lane-24.
  const bool active = (lane < 8) || (lane >= 24);
  const int  k      = (lane < 8) ? lane : (lane - 24);
  float val = acc[0];
  val = (k == 1) ? acc[1] : val;
  val = (k == 2) ? acc[2] : val;
  val = (k == 3) ? acc[3] : val;
  val = (k == 4) ? acc[4] : val;
  val = (k == 5) ? acc[5] : val;
  val = (k == 6) ? acc[6] : val;
  val = (k == 7) ? acc[7] : val;

  if (active) {
    const int eo = e0 + ((lane < 8) ? lane : (lane - 16));
    const int uu = eu[eo], vv = ev[eo];
    const float nu = nrm[(size_t)b * Vn + uu];
    const float nv = nrm[(size_t)b * Vn + vv];
    weight[(size_t)b * NEDGE + eo] = val / (nu * nv);
  }
}

// ------------------------------ Boruvka ------------------------------------

__global__ void k_init_state(int* __restrict__ parent, int* __restrict__ selected) {
  int i = blockIdx.x * 256 + threadIdx.x;
  int b = blockIdx.y;
  if (i < Vn)    parent[(size_t)b * Vn + i] = i;
  if (i < NEDGE) selected[(size_t)b * NEDGE + i] = 0;
}

__global__ void k_compute_root(const int* __restrict__ parent,
                               int* __restrict__ root,
                               unsigned* __restrict__ minkey,
                               int* __restrict__ minidx) {
  int n = blockIdx.x * 256 + threadIdx.x;
  int b = blockIdx.y;
  if (n >= Vn) return;
  const int* par = parent + (size_t)b * Vn;
  int p = par[n];
  // equivalent to parent^(2^17): chase to the self-loop root (chains < V)
  for (int it = 0; it < (1 << ITERS); ++it) {
    int q = par[p];
    if (q == p) break;
    p = q;
  }
  root  [(size_t)b * Vn + n] = p;
  minkey[(size_t)b * Vn + n] = ENC_INF;
  minidx[(size_t)b * Vn + n] = BIGE;
}

__global__ void k_scatter_minw(const int* __restrict__ eu, const int* __restrict__ ev,
                               const int* __restrict__ root,
                               const float* __restrict__ weight,
                               unsigned* __restrict__ minkey) {
  int e = blockIdx.x * 256 + threadIdx.x;
  int b = blockIdx.y;
  if (e >= NEDGE) return;
  const int* rt = root + (size_t)b * Vn;
  int ru = rt[eu[e]], rv = rt[ev[e]];
  if (ru == rv) return;
  unsigned k = fkey(weight[(size_t)b * NEDGE + e]);
  atomicMin(&minkey[(size_t)b * Vn + ru], k);
  atomicMin(&minkey[(size_t)b * Vn + rv], k);
}

__global__ void k_scatter_minidx(const int* __restrict__ eu, const int* __restrict__ ev,
                                 const int* __restrict__ root,
                                 const float* __restrict__ weight,
                                 const unsigned* __restrict__ minkey,
                                 int* __restrict__ minidx) {
  int e = blockIdx.x * 256 + threadIdx.x;
  int b = blockIdx.y;
  if (e >= NEDGE) return;
  const int* rt = root + (size_t)b * Vn;
  int ru = rt[eu[e]], rv = rt[ev[e]];
  if (ru == rv) return;
  float w = weight[(size_t)b * NEDGE + e];
  if (w == fdecode(minkey[(size_t)b * Vn + ru])) atomicMin(&minidx[(size_t)b * Vn + ru], e);
  if (w == fdecode(minkey[(size_t)b * Vn + rv])) atomicMin(&minidx[(size_t)b * Vn + rv], e);
}

__global__ void k_mark_selected(const int* __restrict__ eu, const int* __restrict__ ev,
                                const int* __restrict__ root,
                                const int* __restrict__ minidx,
                                int* __restrict__ selected) {
  int e = blockIdx.x * 256 + threadIdx.x;
  int b = blockIdx.y;
  if (e >= NEDGE) return;
  const int* rt = root + (size_t)b * Vn;
  int ru = rt[eu[e]], rv = rt[ev[e]];
  if (ru == rv) return;
  const int* mi = minidx + (size_t)b * Vn;
  if (mi[ru] == e || mi[rv] == e) selected[(size_t)b * NEDGE + e] = 1;
}

__global__ void k_compute_other(const int* __restrict__ eu, const int* __restrict__ ev,
                                const int* __restrict__ root,
                                const int* __restrict__ minidx,
                                int* __restrict__ other) {
  int n = blockIdx.x * 256 + threadIdx.x;
  int b = blockIdx.y;
  if (n >= Vn) return;
  int mi = minidx[(size_t)b * Vn + n];
  int e  = min(mi, NEDGE - 1);           // JAX OOB-gather clamps to last elem
  const int* rt = root + (size_t)b * Vn;
  int ru = rt[eu[e]], rv = rt[ev[e]];
  other[(size_t)b * Vn + n] = (ru == n) ? rv : ru;
}

__global__ void k_update_parent(const int* __restrict__ root,
                                const int* __restrict__ other,
                                const int* __restrict__ minidx,
                                int* __restrict__ parent) {
  int n = blockIdx.x * 256 + threadIdx.x;
  int b = blockIdx.y;
  if (n >= Vn) return;
  const int* rt = root   + (size_t)b * Vn;
  const int* ot = other  + (size_t)b * Vn;
  const int* mi = minidx + (size_t)b * Vn;
  int m  = rt[n];
  int om = ot[m];
  bool hasm   = mi[m]  < BIGE;
  bool haso   = mi[om] < BIGE;
  bool mutual = hasm && haso && (ot[om] == m);
  bool link   = hasm && !(mutual && (m < om));
  parent[(size_t)b * Vn + n] = link ? om : m;
}

// --------------------------- ordered compaction ----------------------------

__global__ void k_count_sel(const int* __restrict__ selected, int* __restrict__ bcnt) {
  int b   = blockIdx.y;
  int e   = blockIdx.x * 1024 + threadIdx.x;
  int prd = (e < NEDGE) ? selected[(size_t)b * NEDGE + e] : 0;
  unsigned long long bal = __ballot(prd != 0);
  __shared__ int wcnt[32];
  int lane = threadIdx.x & 31, wv = threadIdx.x >> 5;
  if (lane == 0) wcnt[wv] = __popcll(bal);
  __syncthreads();
  if (threadIdx.x == 0) {
    int s = 0;
    for (int i = 0; i < 32; ++i) s += wcnt[i];
    bcnt[(size_t)b * NBC + blockIdx.x] = s;
  }
}

__global__ void k_scan_blocks(const int* __restrict__ bcnt, int* __restrict__ boff) {
  int b = blockIdx.x;
  if (threadIdx.x != 0) return;
  int acc = 0;
  for (int i = 0; i < NBC; ++i) {
    boff[(size_t)b * NBC + i] = acc;
    acc += bcnt[(size_t)b * NBC + i];
  }
}

__global__ void k_fill_out(int* __restrict__ out) {
  int i = blockIdx.x * 256 + threadIdx.x;
  int b = blockIdx.y;
  if (i < NOUT) out[(size_t)b * NOUT + i] = BIGE;
}

__global__ void k_write_sel(const int* __restrict__ selected,
                            const int* __restrict__ boff,
                            int* __restrict__ out) {
  int b   = blockIdx.y;
  int e   = blockIdx.x * 1024 + threadIdx.x;
  int prd = (e < NEDGE) ? selected[(size_t)b * NEDGE + e] : 0;
  unsigned long long bal = __ballot(prd != 0);
  __shared__ int wcnt[32];
  __shared__ int woff[32];
  int lane = threadIdx.x & 31, wv = threadIdx.x >> 5;
  int pre  = __popcll(bal & ((1ull << lane) - 1ull));
  if (lane == 0) wcnt[wv] = __popcll(bal);
  __syncthreads();
  if (threadIdx.x == 0) {
    int a = 0;
    for (int i = 0; i < 32; ++i) { woff[i] = a; a += wcnt[i]; }
  }
  __syncthreads();
  if (prd) {
    int pos = boff[(size_t)b * NBC + blockIdx.x] + woff[wv] + pre;
    if (pos < NOUT) out[(size_t)b * NOUT + pos] = e;
  }
}

}  // namespace

// ------------------------------- launcher ----------------------------------

extern "C" void kernel_launch(void* const* d_in, const int* in_sizes, int n_in,
                              void* d_out, int out_size, void* d_ws, size_t ws_size,
                              hipStream_t stream) {
  (void)in_sizes; (void)n_in; (void)out_size; (void)ws_size;
  const float* guide = (const float*)d_in[0];
  int* out = (int*)d_out;

  char* p = (char*)d_ws;
  auto take = [&p](size_t bytes) -> void* {
    void* r = (void*)p;
    p += (bytes + 255) & ~size_t(255);
    return r;
  };
  int*      d_u    = (int*)     take(sizeof(int)      * NEDGE);
  int*      d_v    = (int*)     take(sizeof(int)      * NEDGE);
  float*    d_nrm  = (float*)   take(sizeof(float)    * Bn * Vn);
  float*    d_w    = (float*)   take(sizeof(float)    * Bn * NEDGE);
  int*      d_par  = (int*)     take(sizeof(int)      * Bn * Vn);
  int*      d_root = (int*)     take(sizeof(int)      * Bn * Vn);
  unsigned* d_mk   = (unsigned*)take(sizeof(unsigned) * Bn * Vn);
  int*      d_mi   = (int*)     take(sizeof(int)      * Bn * Vn);
  int*      d_ot   = (int*)     take(sizeof(int)      * Bn * Vn);
  int*      d_sel  = (int*)     take(sizeof(int)      * Bn * NEDGE);
  int*      d_bc   = (int*)     take(sizeof(int)      * Bn * NBC);
  int*      d_bo   = (int*)     take(sizeof(int)      * Bn * NBC);

  const int gbE   = (NEDGE + 255) / 256;   // 1004
  const int gbV   = Vn / 256;              // 432 (exact)
  const int gbWm  = (NEDGE / 16) / 8;      // 2007 (exact: 16 edges/wave, 8 waves/block)
  const int gbOut = (NOUT + 255) / 256;    // 432

  k_build_index<<<dim3(gbE), 256, 0, stream>>>(d_u, d_v);
  k_node_norm<<<dim3(gbV, Bn), 256, 0, stream>>>(guide, d_nrm);
  k_edge_weight_wmma<<<dim3(gbWm, Bn), 256, 0, stream>>>(guide, d_u, d_v, d_nrm, d_w);
  k_init_state<<<dim3(gbE, Bn), 256, 0, stream>>>(d_par, d_sel);

  for (int it = 0; it < ITERS; ++it) {
    k_compute_root  <<<dim3(gbV, Bn), 256, 0, stream>>>(d_par, d_root, d_mk, d_mi);
    k_scatter_minw  <<<dim3(gbE, Bn), 256, 0, stream>>>(d_u, d_v, d_root, d_w, d_mk);
    k_scatter_minidx<<<dim3(gbE, Bn), 256, 0, stream>>>(d_u, d_v, d_root, d_w, d_mk, d_mi);
    k_mark_selected <<<dim3(gbE, Bn), 256, 0, stream>>>(d_u, d_v, d_root, d_mi, d_sel);
    k_compute_other <<<dim3(gbV, Bn), 256, 0, stream>>>(d_u, d_v, d_root, d_mi, d_ot);
    k_update_parent <<<dim3(gbV, Bn), 256, 0, stream>>>(d_root, d_ot, d_mi, d_par);
  }

  k_count_sel  <<<dim3(NBC, Bn), 1024, 0, stream>>>(d_sel, d_bc);
  k_scan_blocks<<<dim3(Bn),       64, 0, stream>>>(d_bc, d_bo);
  k_fill_out   <<<dim3(gbOut, Bn), 256, 0, stream>>>(out);
  k_write_sel  <<<dim3(NBC, Bn), 1024, 0, stream>>>(d_sel, d_bo, out);
}